// GCN3_49478023250097
// MI455X (gfx1250) — compile-verified
//
#include <hip/hip_runtime.h>
#include <hip/hip_bf16.h>

// ---------------------------------------------------------------------------
// GCN-3 on MI455X: 3x (COO SpMM -> [N,64]@[64,64] GEMM + bias + ReLU),
// fused mean-pool, softmax head.
//   * Dense layers: v_wmma_f32_16x16x4_f32, 4 M-tiles per wave32 (4 indep
//     accumulators, B fragment reused 4x per k-step).
//   * Layer-1 K=5 zero-padded to K=8 in DATA (stride-8 S buffer + padded W1)
//     so the GEMM inner loop is branch-free for all layers.
//   * SpMM: edge-parallel global_atomic_add_f32 (L2-resident: features+edges
//     fit in the 192MB L2).
//   * Pooling: batch is sorted; 8-row tile nearly always maps to one graph ->
//     single atomic of the register-summed contribution (8x fewer atomics).
//   * Epilogue bounds check hoisted to tile level (scalar branch): stores are
//     unconditional clause-able global_store_b32 with constant offsets.
// ---------------------------------------------------------------------------

typedef __attribute__((ext_vector_type(2))) float v2f;
typedef __attribute__((ext_vector_type(8))) float v8f;

#define HID 64
#define NGRAPH 256
#define NOUT 10

// ---------------------------------------------------------------- zero fill
__global__ void zero_f32(float* __restrict__ p, long n) {
    long i = blockIdx.x * (long)blockDim.x + threadIdx.x;
    if (i < n) p[i] = 0.0f;
}

// ------------------------------------------------------- per-graph node count
__global__ void count_kernel(const int* __restrict__ batch,
                             float* __restrict__ cnts, int n) {
    int i = blockIdx.x * blockDim.x + threadIdx.x;
    if (i < n) atomicAdd(&cnts[batch[i]], 1.0f);
}

// ------------------------------------------- pad W1 [5,64] -> Wp [8,64] (zeros)
__global__ void pad_w1(const float* __restrict__ W1, float* __restrict__ Wp) {
    int i = threadIdx.x + blockIdx.x * blockDim.x;   // 512 threads total
    if (i < 8 * HID) Wp[i] = (i < 5 * HID) ? W1[i] : 0.0f;
}

// -------------------------------------------------------------- COO SpMM D=5
// s[row, 0..4] += val * x[col, 0..4]; s row stride 8 (cols 5..7 stay zero)
__global__ void spmm5(const int* __restrict__ rows, const int* __restrict__ cols,
                      const float* __restrict__ vals, const float* __restrict__ x,
                      float* __restrict__ s, long n_edges) {
    long e = blockIdx.x * (long)blockDim.x + threadIdx.x;
    if (e >= n_edges) return;
    int r = rows[e], c = cols[e];
    float v = vals[e];
    const float* __restrict__ xs = x + (size_t)c * 5;
    float* __restrict__ sd = s + (size_t)r * 8;
#pragma unroll
    for (int d = 0; d < 5; ++d) atomicAdd(&sd[d], v * xs[d]);
}

// ------------------------------------------------------------- COO SpMM D=64
// s[row, :] += val * x[col, :]; 4 features per thread, 16 threads per edge
__global__ void spmm64(const int* __restrict__ rows, const int* __restrict__ cols,
                       const float* __restrict__ vals, const float* __restrict__ x,
                       float* __restrict__ s, long n_edges) {
    long tid = blockIdx.x * (long)blockDim.x + threadIdx.x;
    if (tid >= n_edges * 16) return;
    long e  = tid >> 4;
    int  c0 = ((int)tid & 15) * 4;
    int  r  = rows[e], c = cols[e];
    float v = vals[e];
    const float* __restrict__ xs = x + (size_t)c * HID + c0;
    float* __restrict__ sd = s + (size_t)r * HID + c0;
#pragma unroll
    for (int j = 0; j < 4; ++j) atomicAdd(&sd[j], v * xs[j]);
}

// ------------------------------------------------- WMMA GEMM + bias + ReLU
// Xout[N,64] = relu(S[N,K] @ W[K,64] + bias); pooled[batch[m],:] += Xout[m,:]/3.
// Wave32 computes MT 16x16 output tiles (same 16-col slice, MT row tiles):
// B fragment loaded once per k-step, reused across MT independent WMMAs.
// Layouts per ISA 7.12.2:
//   A 16x4 f32 : lane=M(0..15), VGPR{0,1} = K{0,1} (lanes 0-15) / K{2,3} (16-31)
//   B col-major: mirrors A striping with lane=N
//   C/D        : VGPR r -> row r (lanes 0-15) / row r+8 (lanes 16-31), N=lane&15
template <int K, int MT>
__global__ void gemm_bias_relu_pool(const float* __restrict__ S,
                                    const float* __restrict__ W,
                                    const float* __restrict__ bias,
                                    float* __restrict__ Xout,
                                    float* __restrict__ pooled,
                                    const int* __restrict__ batch,
                                    int n_nodes) {
    const int lane  = threadIdx.x;              // 0..31 (wave32)
    const int ntile = threadIdx.y;              // 0..3 -> cols [16*ntile, +16)
    const int tbase = blockIdx.x * MT;          // first row tile of this wave
    const int col   = ntile * 16 + (lane & 15);
    const int khalf = (lane >> 4) * 2;          // 0 or 2

    const float* aptr[MT];
#pragma unroll
    for (int t = 0; t < MT; ++t) {
        int mrow = (tbase + t) * 16 + (lane & 15);
        int m    = (mrow < n_nodes) ? mrow : (n_nodes - 1);   // clamp loads
        aptr[t]  = S + (size_t)m * K;
    }

    v8f c[MT];
#pragma unroll
    for (int t = 0; t < MT; ++t) c[t] = (v8f){};

#pragma unroll
    for (int k0 = 0; k0 < K; k0 += 4) {
        const int ka = k0 + khalf;
        v2f b;
        b.x = W[(size_t)(ka + 0) * HID + col];
        b.y = W[(size_t)(ka + 1) * HID + col];
#pragma unroll
        for (int t = 0; t < MT; ++t) {
            v2f a = *(const v2f*)(aptr[t] + ka);              // 8B aligned
            c[t] = __builtin_amdgcn_wmma_f32_16x16x4_f32(
                       false, a, false, b, (short)0, c[t], false, false);
        }
    }

    const float bv     = bias[col];             // same col for all accumulators
    const int   rowoff = (lane >> 4) * 8;
#pragma unroll
    for (int t = 0; t < MT; ++t) {
        const int tile0 = (tbase + t) * 16;
        const int rbase = tile0 + rowoff;
        if (tile0 + 16 <= n_nodes) {            // wave-uniform (scalar) branch
            // ---- full tile fast path: unconditional stores, combined atomic
            float v[8];
            float vsum = 0.0f;
            float* __restrict__ xrow = Xout + (size_t)rbase * HID + col;
#pragma unroll
            for (int r = 0; r < 8; ++r) {
                float u = c[t][r] + bv;
                u = u > 0.0f ? u : 0.0f;
                v[r] = u;
                vsum += u;
                xrow[(size_t)r * HID] = u;      // constant 256B offsets
            }
            int b0 = batch[rbase];
            int b7 = batch[rbase + 7];
            if (b0 == b7) {                     // sorted batch: common case
                atomicAdd(&pooled[(size_t)b0 * HID + col], vsum * (1.0f / 3.0f));
            } else {
#pragma unroll
                for (int r = 0; r < 8; ++r)
                    atomicAdd(&pooled[(size_t)batch[rbase + r] * HID + col],
                              v[r] * (1.0f / 3.0f));
            }
        } else if (tile0 < n_nodes) {
            // ---- ragged tail tile (never hit when N % 16 == 0)
#pragma unroll
            for (int r = 0; r < 8; ++r) {
                int row = rbase + r;
                if (row < n_nodes) {
                    float u = c[t][r] + bv;
                    u = u > 0.0f ? u : 0.0f;
                    Xout[(size_t)row * HID + col] = u;
                    atomicAdd(&pooled[(size_t)batch[row] * HID + col],
                              u * (1.0f / 3.0f));
                }
            }
        }
    }
}

// ------------------------------------------------- mean-pool finish + head
__global__ void final_head(const float* __restrict__ pooled,
                           const float* __restrict__ cnts,
                           const float* __restrict__ Wl,
                           const float* __restrict__ bl,
                           float* __restrict__ out) {
    int g = threadIdx.x;                        // one thread per graph (256)
    float inv = 1.0f / fmaxf(cnts[g], 1.0f);
    float logits[NOUT];
#pragma unroll
    for (int o = 0; o < NOUT; ++o) logits[o] = bl[o];
    for (int k = 0; k < HID; ++k) {
        float p = pooled[(size_t)g * HID + k] * inv;
#pragma unroll
        for (int o = 0; o < NOUT; ++o) logits[o] += p * Wl[(size_t)k * NOUT + o];
    }
    float mx = logits[0];
#pragma unroll
    for (int o = 1; o < NOUT; ++o) mx = fmaxf(mx, logits[o]);
    float sum = 0.0f;
#pragma unroll
    for (int o = 0; o < NOUT; ++o) { logits[o] = __expf(logits[o] - mx); sum += logits[o]; }
    float rs = 1.0f / sum;
#pragma unroll
    for (int o = 0; o < NOUT; ++o) out[(size_t)g * NOUT + o] = logits[o] * rs;
}

// ---------------------------------------------------------------------------
static inline int div_up(long a, int b) { return (int)((a + b - 1) / b); }

extern "C" void kernel_launch(void* const* d_in, const int* in_sizes, int n_in,
                              void* d_out, int out_size, void* d_ws, size_t ws_size,
                              hipStream_t stream) {
    const float* x     = (const float*)d_in[0];
    const int*   rows  = (const int*)  d_in[1];
    const int*   cols  = (const int*)  d_in[2];
    const float* vals  = (const float*)d_in[3];
    const int*   batch = (const int*)  d_in[4];
    const float* W1 = (const float*)d_in[5];
    const float* b1 = (const float*)d_in[6];
    const float* W2 = (const float*)d_in[7];
    const float* b2 = (const float*)d_in[8];
    const float* W3 = (const float*)d_in[9];
    const float* b3 = (const float*)d_in[10];
    const float* Wl = (const float*)d_in[11];
    const float* bl = (const float*)d_in[12];

    const int  N = in_sizes[0] / 5;             // 100000 nodes
    const long E = in_sizes[1];                 // 3,200,000 edges

    // workspace: s[N,64] | xh[N,64] | pooled[256,64] | cnts[256] | Wp[8,64]
    float* s      = (float*)d_ws;
    float* xh     = s      + (size_t)N * HID;
    float* pooled = xh     + (size_t)N * HID;
    float* cnts   = pooled + (size_t)NGRAPH * HID;
    float* Wp     = cnts   + NGRAPH;

    const int  MT     = 4;                      // row tiles per wave
    const int  mtiles = (N + 15) / 16;
    const int  gemmgx = div_up(mtiles, MT);
    const dim3 gblk(32, 4);                     // 4 wave32's: one per col tile

    // pooled + cnts contiguous: zero both, then count nodes per graph
    { long n = (long)NGRAPH * HID + NGRAPH;
      zero_f32<<<div_up(n, 256), 256, 0, stream>>>(pooled, n); }
    count_kernel<<<div_up(N, 256), 256, 0, stream>>>(batch, cnts, N);
    pad_w1<<<2, 256, 0, stream>>>(W1, Wp);

    // ---- layer 1: s[N,8] = pad(L @ x), xh = relu(s @ Wp + b1), pool += xh/3
    zero_f32<<<div_up((long)N * 8, 256), 256, 0, stream>>>(s, (long)N * 8);
    spmm5<<<div_up(E, 256), 256, 0, stream>>>(rows, cols, vals, x, s, E);
    gemm_bias_relu_pool<8, MT><<<gemmgx, gblk, 0, stream>>>(s, Wp, b1, xh, pooled, batch, N);

    // ---- layer 2: s = L @ xh, xh = relu(s @ W2 + b2), pool += xh/3
    zero_f32<<<div_up((long)N * HID, 256), 256, 0, stream>>>(s, (long)N * HID);
    spmm64<<<div_up(E * 16, 256), 256, 0, stream>>>(rows, cols, vals, xh, s, E);
    gemm_bias_relu_pool<64, MT><<<gemmgx, gblk, 0, stream>>>(s, W2, b2, xh, pooled, batch, N);

    // ---- layer 3
    zero_f32<<<div_up((long)N * HID, 256), 256, 0, stream>>>(s, (long)N * HID);
    spmm64<<<div_up(E * 16, 256), 256, 0, stream>>>(rows, cols, vals, xh, s, E);
    gemm_bias_relu_pool<64, MT><<<gemmgx, gblk, 0, stream>>>(s, W3, b3, xh, pooled, batch, N);

    // ---- mean pool + linear head + softmax
    final_head<<<1, NGRAPH, 0, stream>>>(pooled, cnts, Wl, bl, (float*)d_out);
}